// UMPS_38130719654434
// MI455X (gfx1250) — compile-verified
//
#include <hip/hip_runtime.h>
#include <cstdint>

// UMPS chain contraction for MI455X (gfx1250), fp32 WMMA path.
// Kernel 1: per (batch, segment) workgroup computes the product of 128
//           consecutive 32x32 transfer matrices via V_WMMA_F32_16X16X4_F32.
//           Segment input slab staged to LDS once via the Tensor Data Mover.
// Kernel 2: per-batch combine of segment products with alpha/omega/output_core.

#define B_    64
#define L_    1024
#define FP1   17          // F + 1
#define D_    32          // bond dim
#define O_    8           // output dim
#define SEG   128         // chain steps per workgroup
#define NSEG  (L_ / SEG)  // 8 segments per batch
#define NSEGH (NSEG / 2)  // 4 per half
#define PSTR  34          // even LDS row stride: aligned b64 frags, conflict-free

typedef __attribute__((ext_vector_type(2))) float v2f;
typedef __attribute__((ext_vector_type(8))) float v8f;
typedef __attribute__((ext_vector_type(4))) unsigned int u32x4;
typedef __attribute__((ext_vector_type(4))) int i32x4;
typedef __attribute__((ext_vector_type(8))) int i32x8;

#if defined(__has_builtin)
#if __has_builtin(__builtin_amdgcn_tensor_load_to_lds)
#define HAVE_TDM 1
#endif
#endif

__global__ __launch_bounds__(128) void umps_segment_kernel(
    const float* __restrict__ inputs,   // (B, L, 17)
    const float* __restrict__ core,     // (32, 17, 32)
    float* __restrict__ Pout)           // (B, NSEG, 32, 32)
{
    __shared__ float Mt_lds[D_ * PSTR];     // M transposed: Mt[n][k]
    __shared__ float P_lds[D_ * PSTR];      // running product, row-major
    __shared__ float xseg[SEG * FP1];       // this segment's input slab (8.7 KB)

    const int tid  = threadIdx.x;
    const int lane = tid & 31;
    const int wave = tid >> 5;          // 4 waves
    const int s    = blockIdx.x;        // segment
    const int b    = blockIdx.y;        // batch

    const int   l0 = s * SEG;
    const float* xg = inputs + ((size_t)b * L_ + l0) * FP1;

    // ---- Stage the whole segment's inputs into LDS via the TDM ----
#if defined(HAVE_TDM)
    if (wave == 0) {
        const unsigned long long ga = (unsigned long long)(uintptr_t)xg;
        const unsigned int lds_off  = (unsigned int)(uintptr_t)&xseg[0];
        const unsigned int N = SEG * FP1;   // 2176 fp32 elements, one row
        u32x4 g0;
        g0[0] = 1u;                                    // count=1, user mode
        g0[1] = lds_off;                               // lds_addr
        g0[2] = (unsigned int)(ga & 0xFFFFFFFFu);      // global_addr[31:0]
        g0[3] = (unsigned int)((ga >> 32) & 0x01FFFFFFu) | (2u << 30); // type=2
        i32x8 g1;
        g1[0] = (2 << 16);                             // data_size = 4 bytes
        g1[1] = (int)((N & 0xFFFFu) << 16);            // tensor_dim0[15:0]
        g1[2] = (int)((N >> 16) & 0xFFFFu) | (1 << 16); // dim0[31:16] | tensor_dim1=1
        g1[3] = (int)((N & 0xFFFFu) << 16);            // tile_dim0 = N
        g1[4] = 1;                                     // tile_dim1=1, tile_dim2=0
        g1[5] = (int)N;                                // tensor_dim0_stride[31:0]
        g1[6] = 0;
        g1[7] = 0;
        i32x4 gz = {0, 0, 0, 0};
#if __clang_major__ >= 23
        i32x8 gz8 = {0, 0, 0, 0, 0, 0, 0, 0};
        __builtin_amdgcn_tensor_load_to_lds(g0, g1, gz, gz, gz8, 0);
#else
        __builtin_amdgcn_tensor_load_to_lds(g0, g1, gz, gz, 0);
#endif
        __builtin_amdgcn_s_wait_tensorcnt(0);
    }
#else
    for (int i = tid; i < SEG * FP1; i += 128) xseg[i] = xg[i];
#endif

    // Each lane keeps its slice of `core` in registers: rows c in
    // [wave*8, wave*8+8), column e = lane, all 17 feature slices.
    float creg[8][FP1];
    #pragma unroll
    for (int cr = 0; cr < 8; ++cr) {
        const int c = wave * 8 + cr;
        #pragma unroll
        for (int f = 0; f < FP1; ++f)
            creg[cr][f] = core[((size_t)c * FP1 + f) * D_ + lane];
    }

    // P = identity
    for (int i = tid; i < D_ * D_; i += 128) {
        const int c = i >> 5, e = i & 31;
        P_lds[c * PSTR + e] = (c == e) ? 1.0f : 0.0f;
    }
    __syncthreads();

    // WMMA tile assignment: wave -> output tile (ti, tj) of the 32x32 product.
    const int ti    = wave >> 1;
    const int tj    = wave & 1;
    const int m     = lane & 15;             // row/col within tile
    const int khalf = (lane >> 4) * 2;       // A/B fragment K split per ISA

    for (int l = 0; l < SEG; ++l) {
        // Build M = sum_f x_f * core[:,f,:]; store transposed Mt[e][c],
        // rows c in [wave*8, wave*8+8), column e = lane.
        #pragma unroll
        for (int cr = 0; cr < 8; ++cr) {
            float acc = 0.0f;
            #pragma unroll
            for (int f = 0; f < FP1; ++f)
                acc = fmaf(xseg[l * FP1 + f], creg[cr][f], acc);
            Mt_lds[lane * PSTR + (wave * 8 + cr)] = acc;
        }
        __syncthreads();

        // P_new(ti,tj) = sum_k P[16ti+m, k] * M[k, 16tj+n], K=32 in chunks of 4.
        v8f acc = {0.f, 0.f, 0.f, 0.f, 0.f, 0.f, 0.f, 0.f};
        #pragma unroll
        for (int kk = 0; kk < 8; ++kk) {
            const int k0 = kk * 4 + khalf;   // even -> 8B-aligned b64 loads
            const v2f a  = *(const v2f*)&P_lds[(16 * ti + m) * PSTR + k0];
            const v2f bf = *(const v2f*)&Mt_lds[(16 * tj + m) * PSTR + k0];
            acc = __builtin_amdgcn_wmma_f32_16x16x4_f32(
                      false, a, false, bf, (short)0, acc, false, false);
        }
        __syncthreads();   // all waves done reading old P / Mt

        // Scatter C tile back into P (C layout: VGPR r -> rows r / r+8).
        const int rbase = 16 * ti + ((lane < 16) ? 0 : 8);
        #pragma unroll
        for (int r = 0; r < 8; ++r)
            P_lds[(rbase + r) * PSTR + 16 * tj + m] = acc[r];
        __syncthreads();
    }

    // Write segment product to workspace (dense 32x32 row-major).
    float* Pg = Pout + ((size_t)b * NSEG + s) * (D_ * D_);
    for (int i = tid; i < D_ * D_; i += 128) {
        const int c = i >> 5, e = i & 31;
        Pg[i] = P_lds[c * PSTR + e];
    }
}

__global__ __launch_bounds__(32) void umps_combine_kernel(
    const float* __restrict__ Pws,     // (B, NSEG, 32, 32)
    const float* __restrict__ alpha,   // (32,)
    const float* __restrict__ omega,   // (32,)
    const float* __restrict__ oc,      // (32, 8, 32)
    float* __restrict__ out)           // (B, 8)
{
    __shared__ float vb[D_];
    __shared__ float wb[D_];
    const int lane = threadIdx.x;
    const int b    = blockIdx.x;

    // Left sweep: v = alpha * P_0 * P_1 * ... * P_{NSEGH-1}
    float v = alpha[lane];
    for (int s = 0; s < NSEGH; ++s) {
        vb[lane] = v;
        __syncthreads();
        const float* P = Pws + ((size_t)b * NSEG + s) * (D_ * D_);
        float acc = 0.0f;
        #pragma unroll
        for (int d = 0; d < D_; ++d)
            acc = fmaf(vb[d], P[d * D_ + lane], acc);
        __syncthreads();
        v = acc;
    }

    // Right sweep: w = P_{NSEGH} * ... * P_{NSEG-1} * omega (right-to-left)
    float w = omega[lane];
    for (int s = NSEG - 1; s >= NSEGH; --s) {
        wb[lane] = w;
        __syncthreads();
        const float* P = Pws + ((size_t)b * NSEG + s) * (D_ * D_);
        float acc = 0.0f;
        #pragma unroll
        for (int e = 0; e < D_; ++e)
            acc = fmaf(P[lane * D_ + e], wb[e], acc);
        __syncthreads();
        w = acc;
    }

    vb[lane] = v;
    wb[lane] = w;
    __syncthreads();

    // out[b,o] = sum_{d,e} v_d * oc[d,o,e] * w_e
    if (lane < O_) {
        float acc = 0.0f;
        for (int d = 0; d < D_; ++d) {
            float t = 0.0f;
            #pragma unroll
            for (int e = 0; e < D_; ++e)
                t = fmaf(oc[((size_t)d * O_ + lane) * D_ + e], wb[e], t);
            acc = fmaf(vb[d], t, acc);
        }
        out[(size_t)b * O_ + lane] = acc;
    }
}

extern "C" void kernel_launch(void* const* d_in, const int* in_sizes, int n_in,
                              void* d_out, int out_size, void* d_ws, size_t ws_size,
                              hipStream_t stream) {
    (void)in_sizes; (void)n_in; (void)out_size; (void)ws_size;
    const float* inputs = (const float*)d_in[0];   // (64, 1024, 17)
    const float* core   = (const float*)d_in[1];   // (32, 17, 32)
    const float* alpha  = (const float*)d_in[2];   // (32,)
    const float* omega  = (const float*)d_in[3];   // (32,)
    const float* oc     = (const float*)d_in[4];   // (32, 8, 32)
    float* out = (float*)d_out;                    // (64, 8)
    float* Pws = (float*)d_ws;                     // B*NSEG*32*32 floats = 2 MB

    dim3 grid1(NSEG, B_);
    umps_segment_kernel<<<grid1, 128, 0, stream>>>(inputs, core, Pws);
    umps_combine_kernel<<<B_, 32, 0, stream>>>(Pws, alpha, omega, oc, out);
}